// Topkpool_8478265442581
// MI455X (gfx1250) — compile-verified
//
#include <hip/hip_runtime.h>
#include <hip/hip_bf16.h>
#include <math.h>

#define BB  128
#define NN  512
#define CC  128
#define EE  (BB * NN * 8)        // 524288
#define KK1 410
#define KK2 328
#define KK3 263

typedef __attribute__((ext_vector_type(2))) float v2f;
typedef __attribute__((ext_vector_type(8))) float v8f;

// ---------------- zero (float4 granularity) ----------------
__global__ void k_zero(float* __restrict__ p, long long n4) {
  long long i = (long long)blockIdx.x * blockDim.x + threadIdx.x;
  if (i < n4) ((float4*)p)[i] = make_float4(0.f, 0.f, 0.f, 0.f);
}

// ---------------- edge gather + segment-sum (scatter-add) ----------------
// one thread per (edge, float4 chunk); agg is L2-resident so f32 atomics are cheap
__global__ void k_scatter(const float* __restrict__ x,
                          const int* __restrict__ src,
                          const int* __restrict__ dst,
                          const float* __restrict__ mask,   // nullptr => all ones
                          float* __restrict__ agg) {
  long long t = (long long)blockIdx.x * blockDim.x + threadIdx.x;
  int e = (int)(t >> 5);
  int q = (int)(t & 31);
  if (e >= EE) return;
  float m = mask ? mask[e] : 1.0f;
  if (m == 0.0f) return;
  float4 v = ((const float4*)(x + (long long)src[e] * CC))[q];
  float* d = agg + (long long)dst[e] * CC + (q << 2);
  atomicAdd(d + 0, v.x * m);
  atomicAdd(d + 1, v.y * m);
  atomicAdd(d + 2, v.z * m);
  atomicAdd(d + 3, v.w * m);
}

// ---------------- fused dual GEMM: Out = relu(Agg@Wrel + X@Wroot + brel) ------
// fp32 WMMA 16x16x4; one wave per 16x16 tile; grid = M/16 blocks of 256 (8 waves)
// A frag (16x4 f32): lanes 0-15 rows M0..15, v0=K{0|2}, v1=K{1|3} (koff by half-wave)
// B frag (4x16 f32): lanes 0-15 cols N0..15, same K split
// C/D   (16x16 f32): VGPR i -> M=i (lanes 0-15) / M=8+i (lanes 16-31), N=lane&15
__global__ void k_conv_gemm(const float* __restrict__ Agg,
                            const float* __restrict__ X,
                            const float* __restrict__ Wrel,
                            const float* __restrict__ Wroot,
                            const float* __restrict__ brel,
                            float* __restrict__ Out) {
  int wave  = (blockIdx.x << 3) | (threadIdx.x >> 5);
  int lane  = threadIdx.x & 31;
  int mTile = wave >> 3;              // CC/16 = 8 column tiles
  int nTile = wave & 7;
  int row   = (mTile << 4) + (lane & 15);
  int koff  = (lane >> 4) << 1;       // 0 or 2
  int col   = (nTile << 4) + (lane & 15);
  const float* aRel  = Agg + (long long)row * CC;
  const float* aRoot = X   + (long long)row * CC;
  v8f acc = {};
  for (int k0 = 0; k0 < CC; k0 += 4) {
    v2f a1 = *(const v2f*)(aRel  + k0 + koff);   // 8B-aligned vector load
    v2f a2 = *(const v2f*)(aRoot + k0 + koff);
    v2f b1, b2;
    b1.x = Wrel [(k0 + koff)     * CC + col];
    b1.y = Wrel [(k0 + koff + 1) * CC + col];
    b2.x = Wroot[(k0 + koff)     * CC + col];
    b2.y = Wroot[(k0 + koff + 1) * CC + col];
    acc = __builtin_amdgcn_wmma_f32_16x16x4_f32(false, a1, false, b1,
                                                (short)0, acc, false, false);
    acc = __builtin_amdgcn_wmma_f32_16x16x4_f32(false, a2, false, b2,
                                                (short)0, acc, false, false);
  }
  float bias  = brel[col];
  int   rbase = (mTile << 4) + ((lane >> 4) << 3);
  float* obase = Out + (long long)rbase * CC + col;
#pragma unroll
  for (int i = 0; i < 8; ++i) {
    float v = acc[i] + bias;
    obase[i * CC] = v > 0.f ? v : 0.f;
  }
}

// ---------------- per-node score = tanh(h.p / ||p||); also inv = -1 ----------
__global__ void k_score(const float* __restrict__ h,
                        const float* __restrict__ p,
                        float* __restrict__ score,
                        int* __restrict__ inv) {
  int node = (blockIdx.x << 3) | (threadIdx.x >> 5);
  int lane = threadIdx.x & 31;
  const float* hr = h + (long long)node * CC;
  float hp = 0.f, pp = 0.f;
#pragma unroll
  for (int i = 0; i < 4; ++i) {
    int c = lane + (i << 5);
    float pv = p[c];
    hp += hr[c] * pv;
    pp += pv * pv;
  }
#pragma unroll
  for (int off = 16; off >= 1; off >>= 1) {   // wave32 reduction
    hp += __shfl_xor(hp, off, 32);
    pp += __shfl_xor(pp, off, 32);
  }
  if (lane == 0) {
    score[node] = tanhf(hp / sqrtf(pp));
    inv[node]   = -1;
  }
}

// ---------------- per-graph bitonic top-k (n <= 512) -------------------------
__global__ void k_topk(const float* __restrict__ score,
                       int n, int k,
                       int* __restrict__ perm,
                       float* __restrict__ vals,
                       int* __restrict__ inv) {
  __shared__ float sv[512];
  __shared__ int   si[512];
  int g = blockIdx.x;
  int t = threadIdx.x;                 // 512 threads
  if (t < n) { sv[t] = score[g * n + t]; si[t] = t; }
  else       { sv[t] = -__builtin_inff(); si[t] = t; }
  __syncthreads();
  for (int size = 2; size <= 512; size <<= 1) {
    for (int stride = size >> 1; stride > 0; stride >>= 1) {
      int partner = t ^ stride;
      if (partner > t) {
        bool desc = ((t & size) == 0);
        float v1 = sv[t], v2 = sv[partner];
        int   i1 = si[t], i2 = si[partner];
        bool sw = desc ? (v2 > v1 || (v2 == v1 && i2 < i1))
                       : (v2 < v1 || (v2 == v1 && i2 > i1));
        if (sw) { sv[t] = v2; sv[partner] = v1; si[t] = i2; si[partner] = i1; }
      }
      __syncthreads();
    }
  }
  if (t < k) {
    int orig = si[t];
    perm[g * k + t] = g * n + orig;
    vals[g * k + t] = sv[t];
    inv[g * n + orig] = g * k + t;
  }
}

// ---------------- pooled gather * score ----------------
__global__ void k_pool(const float* __restrict__ h,
                       const int* __restrict__ perm,
                       const float* __restrict__ vals,
                       float* __restrict__ out, int m) {
  long long t = (long long)blockIdx.x * blockDim.x + threadIdx.x;
  int i = (int)(t >> 5);
  int q = (int)(t & 31);
  if (i >= m) return;
  float s = vals[i];
  float4 v = ((const float4*)(h + (long long)perm[i] * CC))[q];
  ((float4*)(out + (long long)i * CC))[q] =
      make_float4(v.x * s, v.y * s, v.z * s, v.w * s);
}

// ---------------- edge index remap + mask ----------------
__global__ void k_remap(const int* __restrict__ src_in,
                        const int* __restrict__ dst_in,
                        const float* __restrict__ mask_in,  // nullptr => ones
                        const int* __restrict__ inv,
                        int* __restrict__ src_out,
                        int* __restrict__ dst_out,
                        float* __restrict__ mask_out) {
  int e = blockIdx.x * blockDim.x + threadIdx.x;
  if (e >= EE) return;
  int ns = inv[src_in[e]];
  int nd = inv[dst_in[e]];
  float m = mask_in ? mask_in[e] : 1.0f;
  mask_out[e] = (ns >= 0 && nd >= 0) ? m : 0.0f;
  src_out[e] = ns > 0 ? ns : 0;
  dst_out[e] = nd > 0 ? nd : 0;
}

// ---------------- readout: gmp || gap, accumulated into g --------------------
__global__ void k_readout(const float* __restrict__ h, int k,
                          float* __restrict__ g, int accumulate) {
  int gr = blockIdx.x;
  int c  = threadIdx.x;                // 128 threads, coalesced per row
  const float* base = h + (long long)gr * k * CC + c;
  float mx = -__builtin_inff(), sm = 0.f;
  for (int r = 0; r < k; ++r) {
    float v = base[(long long)r * CC];
    mx = fmaxf(mx, v);
    sm += v;
  }
  float mean = sm / (float)k;
  if (accumulate) {
    g[gr * 256 + c]       += mx;
    g[gr * 256 + 128 + c] += mean;
  } else {
    g[gr * 256 + c]       = mx;
    g[gr * 256 + 128 + c] = mean;
  }
}

// ---------------- MLP + log_softmax, one block per graph ---------------------
__global__ void k_mlp(const float* __restrict__ g,
                      const float* __restrict__ W1, const float* __restrict__ b1,
                      const float* __restrict__ W2, const float* __restrict__ b2,
                      const float* __restrict__ W3, const float* __restrict__ b3,
                      float* __restrict__ out) {
  __shared__ float gin[256];
  __shared__ float h1[128];
  __shared__ float h2[64];
  __shared__ float lg[10];
  int r = blockIdx.x;
  int t = threadIdx.x;                 // 256 threads
  gin[t] = g[r * 256 + t];
  __syncthreads();
  if (t < 128) {
    float acc = b1[t];
    for (int kk = 0; kk < 256; ++kk) acc += gin[kk] * W1[kk * 128 + t];
    h1[t] = fmaxf(acc, 0.f);
  }
  __syncthreads();
  if (t < 64) {
    float acc = b2[t];
    for (int kk = 0; kk < 128; ++kk) acc += h1[kk] * W2[kk * 64 + t];
    h2[t] = fmaxf(acc, 0.f);
  }
  __syncthreads();
  if (t < 10) {
    float acc = b3[t];
    for (int kk = 0; kk < 64; ++kk) acc += h2[kk] * W3[kk * 10 + t];
    lg[t] = acc;
  }
  __syncthreads();
  if (t == 0) {
    float mx = lg[0];
    for (int i = 1; i < 10; ++i) mx = fmaxf(mx, lg[i]);
    float s = 0.f;
    for (int i = 0; i < 10; ++i) s += expf(lg[i] - mx);
    float lse = mx + logf(s);
    for (int i = 0; i < 10; ++i) out[r * 10 + i] = lg[i] - lse;
  }
}

extern "C" void kernel_launch(void* const* d_in, const int* in_sizes, int n_in,
                              void* d_out, int out_size, void* d_ws, size_t ws_size,
                              hipStream_t stream) {
  (void)in_sizes; (void)n_in; (void)out_size; (void)ws_size;
  const float* x        = (const float*)d_in[0];
  const int*   edge_src = (const int*)d_in[1];
  const int*   edge_dst = (const int*)d_in[2];
  const float* Wroot1 = (const float*)d_in[3];
  const float* Wrel1  = (const float*)d_in[4];
  const float* brel1  = (const float*)d_in[5];
  const float* p1     = (const float*)d_in[6];
  const float* Wroot2 = (const float*)d_in[7];
  const float* Wrel2  = (const float*)d_in[8];
  const float* brel2  = (const float*)d_in[9];
  const float* p2     = (const float*)d_in[10];
  const float* Wroot3 = (const float*)d_in[11];
  const float* Wrel3  = (const float*)d_in[12];
  const float* brel3  = (const float*)d_in[13];
  const float* p3     = (const float*)d_in[14];
  const float* W1 = (const float*)d_in[15];
  const float* b1 = (const float*)d_in[16];
  const float* W2 = (const float*)d_in[17];
  const float* b2 = (const float*)d_in[18];
  const float* W3 = (const float*)d_in[19];
  const float* b3 = (const float*)d_in[20];
  float* out = (float*)d_out;

  const int NN0 = BB * NN;   // 65536
  const int NN1 = BB * KK1;  // 52480
  const int NN2 = BB * KK2;  // 41984
  const int NN3 = BB * KK3;  // 33664

  // workspace carve-out
  char* ws = (char*)d_ws;
  size_t off = 0;
  auto take = [&](size_t bytes) -> void* {
    void* p = ws + off;
    off += (bytes + 255) & ~(size_t)255;
    return p;
  };
  float* AGG   = (float*)take((size_t)NN0 * CC * 4);
  float* H     = (float*)take((size_t)NN0 * CC * 4);
  float* XP    = (float*)take((size_t)NN1 * CC * 4);
  float* SCORE = (float*)take((size_t)NN0 * 4);
  int*   PERM  = (int*)  take((size_t)NN1 * 4);
  float* VALS  = (float*)take((size_t)NN1 * 4);
  int*   INV   = (int*)  take((size_t)NN0 * 4);
  int*   SRCA  = (int*)  take((size_t)EE * 4);
  int*   DSTA  = (int*)  take((size_t)EE * 4);
  float* MASKA = (float*)take((size_t)EE * 4);
  int*   SRCB  = (int*)  take((size_t)EE * 4);
  int*   DSTB  = (int*)  take((size_t)EE * 4);
  float* MASKB = (float*)take((size_t)EE * 4);
  float* G     = (float*)take((size_t)BB * 256 * 4);

  const int scatterBlocks = EE / 8;         // (E*32 threads)/256

  // ---------------- layer 1 ----------------
  {
    long long n4 = (long long)NN0 * CC / 4;
    k_zero<<<(unsigned)((n4 + 255) / 256), 256, 0, stream>>>(AGG, n4);
    k_scatter<<<scatterBlocks, 256, 0, stream>>>(x, edge_src, edge_dst, nullptr, AGG);
    k_conv_gemm<<<NN0 / 16, 256, 0, stream>>>(AGG, x, Wrel1, Wroot1, brel1, H);
    k_score<<<NN0 / 8, 256, 0, stream>>>(H, p1, SCORE, INV);
    k_topk<<<BB, 512, 0, stream>>>(SCORE, NN, KK1, PERM, VALS, INV);
    k_pool<<<NN1 / 8, 256, 0, stream>>>(H, PERM, VALS, XP, NN1);
    k_readout<<<BB, 128, 0, stream>>>(XP, KK1, G, 0);
    k_remap<<<EE / 256, 256, 0, stream>>>(edge_src, edge_dst, nullptr, INV,
                                          SRCA, DSTA, MASKA);
  }
  // ---------------- layer 2 ----------------
  {
    long long n4 = (long long)NN1 * CC / 4;
    k_zero<<<(unsigned)((n4 + 255) / 256), 256, 0, stream>>>(AGG, n4);
    k_scatter<<<scatterBlocks, 256, 0, stream>>>(XP, SRCA, DSTA, MASKA, AGG);
    k_conv_gemm<<<NN1 / 16, 256, 0, stream>>>(AGG, XP, Wrel2, Wroot2, brel2, H);
    k_score<<<NN1 / 8, 256, 0, stream>>>(H, p2, SCORE, INV);
    k_topk<<<BB, 512, 0, stream>>>(SCORE, KK1, KK2, PERM, VALS, INV);
    k_pool<<<NN2 / 8, 256, 0, stream>>>(H, PERM, VALS, XP, NN2);
    k_readout<<<BB, 128, 0, stream>>>(XP, KK2, G, 1);
    k_remap<<<EE / 256, 256, 0, stream>>>(SRCA, DSTA, MASKA, INV,
                                          SRCB, DSTB, MASKB);
  }
  // ---------------- layer 3 ----------------
  {
    long long n4 = (long long)NN2 * CC / 4;
    k_zero<<<(unsigned)((n4 + 255) / 256), 256, 0, stream>>>(AGG, n4);
    k_scatter<<<scatterBlocks, 256, 0, stream>>>(XP, SRCB, DSTB, MASKB, AGG);
    k_conv_gemm<<<NN2 / 16, 256, 0, stream>>>(AGG, XP, Wrel3, Wroot3, brel3, H);
    k_score<<<NN2 / 8, 256, 0, stream>>>(H, p3, SCORE, INV);
    k_topk<<<BB, 512, 0, stream>>>(SCORE, KK2, KK3, PERM, VALS, INV);
    k_pool<<<NN3 / 8, 256, 0, stream>>>(H, PERM, VALS, XP, NN3);
    k_readout<<<BB, 128, 0, stream>>>(XP, KK3, G, 1);
  }
  // ---------------- classifier ----------------
  k_mlp<<<BB, 256, 0, stream>>>(G, W1, b1, W2, b2, W3, b3, out);
}